// EnvAttention_3298534883833
// MI455X (gfx1250) — compile-verified
//
#include <hip/hip_runtime.h>
#include <math.h>

typedef __attribute__((ext_vector_type(2))) float v2f;
typedef __attribute__((ext_vector_type(4))) float f4v;
typedef __attribute__((ext_vector_type(8))) float v8f;
typedef __attribute__((ext_vector_type(4))) int   i4v;

#define AS1 __attribute__((address_space(1)))
#define AS3 __attribute__((address_space(3)))

#if __has_builtin(__builtin_amdgcn_global_load_async_to_lds_b128)
#define USE_ASYNC 1
#else
#define USE_ASYNC 0
#endif

#define NTOK   131072
#define SEGS   128
#define HEADS  8
#define KD     64
#define ROWF   1024              // floats per kv row = H*2*K
#define WAVES  4                 // waves per block (128 threads)
#define TILE   16
#define LDS_ROW 132              // 128 payload + 4 pad floats (bank-conflict-free)
#define WAVE_STAGE (TILE * LDS_ROW)            // 2112 floats per buffer
#define COMBINE_BASE (WAVES * 2 * WAVE_STAGE)  // 16896 floats (double-buffered)
#define COMBINE_STRIDE 66                      // m, den, acc[64]

__device__ __forceinline__ float bcast_lane(float x, int lane) {
  return __int_as_float(__builtin_amdgcn_readlane(__float_as_int(x), lane));
}

// Issue the 16 async row-copies of one FULL tile into `stage` (no waits).
__device__ __forceinline__ void issue_rows(const float* __restrict__ kv,
                                           float* __restrict__ stage,
                                           int base, int h, int lane) {
  const float* gp = kv + (size_t)base * ROWF + h * (2 * KD) + lane * 4;
  float* lp = stage + lane * 4;
#if USE_ASYNC
#pragma unroll
  for (int r = 0; r < TILE; ++r)
    __builtin_amdgcn_global_load_async_to_lds_b128(
        (AS1 i4v*)(gp + (size_t)r * ROWF),
        (AS3 i4v*)(lp + r * LDS_ROW), 0, 0);
#else
  f4v tmp[TILE];
#pragma unroll
  for (int r = 0; r < TILE; ++r)
    tmp[r] = *(const f4v*)(gp + (size_t)r * ROWF);
#pragma unroll
  for (int r = 0; r < TILE; ++r)
    *(f4v*)(lp + r * LDS_ROW) = tmp[r];
#endif
}

// Consume one staged tile: WMMA scores -> online-softmax update.
template <bool MASKED>
__device__ __forceinline__ void process_tile(const float* __restrict__ stage,
                                             const float* bx, const float* by,
                                             int base, int row1,
                                             int lane, int koff,
                                             float& m, float& den,
                                             float& acc0, float& acc1) {
  // covers fallback ds-stores and masked-tail zero-fill (free otherwise)
  asm volatile("s_wait_dscnt 0x0" ::: "memory");

  // ---- batch all 16 A-operand chunks, then run the WMMA chain stall-free ----
  v2f a[16];
#pragma unroll
  for (int kk = 0; kk < 16; ++kk)
    a[kk] = *(const v2f*)(stage + (lane & 15) * LDS_ROW + kk * 4 + koff);

  v8f c = {0.f, 0.f, 0.f, 0.f, 0.f, 0.f, 0.f, 0.f};
#pragma unroll
  for (int kk = 0; kk < 16; ++kk) {
    v2f b = { bx[kk], by[kk] };
    c = __builtin_amdgcn_wmma_f32_16x16x4_f32(false, a[kk], false, b,
                                              (short)0, c, false, false);
  }

  // ---- broadcast the 16 row-scores to every lane (v_readlane -> SGPR) ----
  float sc[16];
#pragma unroll
  for (int j = 0; j < 8; ++j) {
    sc[j]     = bcast_lane(c[j], 0);    // D vgpr j, lanes 0-15  -> M = j
    sc[j + 8] = bcast_lane(c[j], 16);   // D vgpr j, lanes 16-31 -> M = j+8
  }
  if (MASKED) {
    const int nvalid = row1 - base;     // 1..15 in the tail tile
#pragma unroll
    for (int r = 0; r < TILE; ++r)
      if (r >= nvalid) sc[r] = -INFINITY;
  }

  // ---- online softmax update (per-wave running state) ----
  float tm = sc[0];
#pragma unroll
  for (int r = 1; r < TILE; ++r) tm = fmaxf(tm, sc[r]);
  const float mnew  = fmaxf(m, tm);
  const float alpha = __expf(m - mnew);         // m=-inf first tile -> 0
  den *= alpha; acc0 *= alpha; acc1 *= alpha;
#pragma unroll
  for (int r = 0; r < TILE; ++r) {
    const float e = __expf(sc[r] - mnew);
    const v2f vv  = *(const v2f*)(stage + r * LDS_ROW + KD + lane * 2);
    den += e;
    acc0 = fmaf(e, vv[0], acc0);
    acc1 = fmaf(e, vv[1], acc1);
  }
  m = mnew;
}

__global__ void __launch_bounds__(128)
env_attn_kernel(const float* __restrict__ kv,
                const int*   __restrict__ seg,
                const float* __restrict__ q,
                const float* __restrict__ sp,
                float*       __restrict__ out) {
  extern __shared__ float smem[];
  const int bid  = blockIdx.x;
  const int sidx = bid >> 3;          // segment
  const int h    = bid & 7;           // head
  const int tid  = (int)threadIdx.x;
  const int wave = tid >> 5;
  const int lane = tid & 31;

  // ---- binary-search contiguous segment bounds (seg_ids ascending) ----
  int lo = 0, hi = NTOK;
  while (lo < hi) { int m_ = (lo + hi) >> 1; if (seg[m_] < sidx) lo = m_ + 1; else hi = m_; }
  const int row0 = lo;
  lo = 0; hi = NTOK;
  while (lo < hi) { int m_ = (lo + hi) >> 1; if (seg[m_] <= sidx) lo = m_ + 1; else hi = m_; }
  const int row1 = lo;

  // ---- envelope*scale folded into B operand; preload per-lane B chunks ----
  const float f = (fabsf(sp[0]) + 1.0f) * 0.125f;   // (|s|+1) / sqrt(64)
  const int koff = (lane >= 16) ? 2 : 0;            // wave32 16x4 A/B layout half
  float bx[16], by[16];
#pragma unroll
  for (int kk = 0; kk < 16; ++kk) {
    bx[kk] = q[h * KD + kk * 4 + koff]     * f;
    by[kk] = q[h * KD + kk * 4 + koff + 1] * f;
  }

  float* stage = smem + wave * (2 * WAVE_STAGE);   // ping-pong pair
  float m = -INFINITY, den = 0.0f, acc0 = 0.0f, acc1 = 0.0f;

  const int len  = row1 - row0;
  const int full = len >> 4;          // number of complete 16-row tiles

  // ---- hot loop: double-buffered; tile t+WAVES loads overlap tile t compute ----
  int buf = 0;
  if (wave < full)
    issue_rows(kv, stage, row0 + wave * TILE, h, lane);
  for (int t = wave; t < full; t += WAVES) {
    const int tn = t + WAVES;
    if (tn < full) {
      issue_rows(kv, stage + (buf ^ 1) * WAVE_STAGE, row0 + tn * TILE, h, lane);
      asm volatile("s_wait_asynccnt 0x10" ::: "memory");  // current tile done
    } else {
      asm volatile("s_wait_asynccnt 0x0" ::: "memory");
    }
    process_tile<false>(stage + buf * WAVE_STAGE, bx, by, row0 + t * TILE,
                        row1, lane, koff, m, den, acc0, acc1);
    buf ^= 1;
  }

  // ---- at most one partial tail tile per segment; its owning wave handles it ----
  if ((len & 15) && ((full & (WAVES - 1)) == wave)) {
    const int base = row0 + full * TILE;
    const float* gp = kv + (size_t)base * ROWF + h * (2 * KD) + lane * 4;
    float* lp = stage + lane * 4;
#pragma unroll
    for (int r = 0; r < TILE; ++r) {
      if (base + r < row1) {
#if USE_ASYNC
        __builtin_amdgcn_global_load_async_to_lds_b128(
            (AS1 i4v*)(gp + (size_t)r * ROWF),
            (AS3 i4v*)(lp + r * LDS_ROW), 0, 0);
#else
        *(f4v*)(lp + r * LDS_ROW) = *(const f4v*)(gp + (size_t)r * ROWF);
#endif
      } else {
        *(f4v*)(lp + r * LDS_ROW) = f4v{0.f, 0.f, 0.f, 0.f};
      }
    }
    asm volatile("s_wait_asynccnt 0x0" ::: "memory");
    process_tile<true>(stage, bx, by, base, row1, lane, koff,
                       m, den, acc0, acc1);
  }

  // ---- merge the per-wave partials ----
  float* comb = smem + COMBINE_BASE + wave * COMBINE_STRIDE;
  if (lane == 0) { comb[0] = m; comb[1] = den; }
  comb[2 + lane * 2]     = acc0;
  comb[2 + lane * 2 + 1] = acc1;
  __syncthreads();

  if (wave == 0) {
    float M = -INFINITY;
#pragma unroll
    for (int w2 = 0; w2 < WAVES; ++w2)
      M = fmaxf(M, smem[COMBINE_BASE + w2 * COMBINE_STRIDE]);
    float D = 0.f, n0 = 0.f, n1 = 0.f;
#pragma unroll
    for (int w2 = 0; w2 < WAVES; ++w2) {
      const float* cw = smem + COMBINE_BASE + w2 * COMBINE_STRIDE;
      const float mw = cw[0];
      const float g  = (mw == -INFINITY) ? 0.f : __expf(mw - M);
      D  = fmaf(cw[1],                g, D);
      n0 = fmaf(cw[2 + lane * 2],     g, n0);
      n1 = fmaf(cw[2 + lane * 2 + 1], g, n1);
    }
    const float inv = 1.0f / D;
    float* op = out + (size_t)sidx * (HEADS * KD) + h * KD + lane * 2;
    op[0] = n0 * inv;
    op[1] = n1 * inv;
  }
}

extern "C" void kernel_launch(void* const* d_in, const int* in_sizes, int n_in,
                              void* d_out, int out_size, void* d_ws, size_t ws_size,
                              hipStream_t stream) {
  (void)in_sizes; (void)n_in; (void)out_size; (void)d_ws; (void)ws_size;
  const float* kv  = (const float*)d_in[0];
  const int*   seg = (const int*)d_in[1];
  const float* q   = (const float*)d_in[2];
  const float* s   = (const float*)d_in[3];
  float* out = (float*)d_out;
  const size_t shmem =
      (size_t)(COMBINE_BASE + WAVES * COMBINE_STRIDE) * sizeof(float); // ~68.6 KB
  env_attn_kernel<<<dim3(SEGS * HEADS), dim3(128), shmem, stream>>>(kv, seg, q, s, out);
}